// GCNConv_25237227831551
// MI455X (gfx1250) — compile-verified
//
#include <hip/hip_runtime.h>

#define D 128

typedef float v2f __attribute__((ext_vector_type(2)));
typedef float v8f __attribute__((ext_vector_type(8)));

// ---------------------------------------------------------------------------
// 1) zero workspace (agg + deg) -- must run every call (atomics accumulate)
// ---------------------------------------------------------------------------
__global__ __launch_bounds__(256) void zero_ws(float* __restrict__ p, int n) {
    int i = blockIdx.x * 256 + threadIdx.x;
    if (i < n) p[i] = 0.0f;
}

// ---------------------------------------------------------------------------
// 2) out-degree via float atomics (exact for counts < 2^24; mean deg ~32)
// ---------------------------------------------------------------------------
__global__ __launch_bounds__(256) void degree_count(const int* __restrict__ erow,
                                                    float* __restrict__ deg, int ne) {
    int i = blockIdx.x * 256 + threadIdx.x;
    if (i < ne) atomicAdd(deg + erow[i], 1.0f);
}

// ---------------------------------------------------------------------------
// 3) deg^-0.5 in place (0 -> +inf, matching the reference's 0**-0.5)
// ---------------------------------------------------------------------------
__global__ __launch_bounds__(256) void deg_rsqrt(float* __restrict__ deg, int n) {
    int i = blockIdx.x * 256 + threadIdx.x;
    if (i < n) deg[i] = rsqrtf(deg[i]);
}

// ---------------------------------------------------------------------------
// 4) edge scatter: one wave32 per edge. Edge id is wave-uniform (readfirstlane)
//    so row/col/weight loads scalarize; each lane moves 4 consecutive floats
//    (global_load_b128 gather + 4x global_atomic_add_f32). x and agg both fit
//    in the 192MB L2, so these atomics stay on-chip.
// ---------------------------------------------------------------------------
__global__ __launch_bounds__(256) void edge_aggregate(
    const float* __restrict__ x, const int* __restrict__ erow,
    const int* __restrict__ ecol, const float* __restrict__ dis,
    float* __restrict__ agg, int ne)
{
    int wid = blockIdx.x * 8 + (threadIdx.x >> 5);
    if (wid >= ne) return;                       // whole-wave uniform exit
    int e   = __builtin_amdgcn_readfirstlane(wid);
    int row = erow[e];
    int col = ecol[e];
    float w = dis[row] * dis[col];
    int lane = threadIdx.x & 31;

    float4 xv = *((const float4*)(x + (size_t)col * D) + lane);
    float* dst = agg + (size_t)row * D + lane * 4;
    atomicAdd(dst + 0, w * xv.x);
    atomicAdd(dst + 1, w * xv.y);
    atomicAdd(dst + 2, w * xv.z);
    atomicAdd(dst + 3, w * xv.w);
}

// ---------------------------------------------------------------------------
// 5) out = agg @ W + bias with V_WMMA_F32_16X16X4_F32.
//    One wave -> one 16x16 output tile; 8 waves per block cover the 8 column
//    tiles of one 16-row strip (A fragments shared via L1).
//    ISA layouts (CDNA5 ISA):
//      A 16x4:  lanes 0-15 hold (M=lane, K=k..k+1), lanes 16-31 (M=lane-16, K=k+2..k+3)
//      B 4x16:  lanes 0-15 hold (K=k..k+1, N=lane), lanes 16-31 (K=k+2..k+3, N=lane-16)
//      C 16x16: VGPR r, lanes 0-15 -> (M=r, N=lane); lanes 16-31 -> (M=r+8, N=lane-16)
//    EXEC must be all 1s: grid is exact (50000 % 16 == 0), no divergence.
// ---------------------------------------------------------------------------
__global__ __launch_bounds__(256) void gemm_bias_wmma(
    const float* __restrict__ agg, const float* __restrict__ Wt,
    const float* __restrict__ bias, float* __restrict__ out)
{
    const int lane = threadIdx.x & 31;
    const int wave = threadIdx.x >> 5;   // 0..7 -> column tile
    const int m0   = blockIdx.x * 16;
    const int n0   = wave * 16;
    const int half = lane >> 4;          // 0: K pair (k,k+1), 1: (k+2,k+3)
    const int l15  = lane & 15;

    // Fold bias into the accumulator: C element (r, lane) maps to column n0+l15.
    float bval = bias[n0 + l15];
    v8f c;
#pragma unroll
    for (int r = 0; r < 8; ++r) c[r] = bval;

    const float* arow = agg + (size_t)(m0 + l15) * D;   // row for A fragment
#pragma unroll 8
    for (int k = 0; k < D; k += 4) {
        v2f a = *(const v2f*)(arow + k + half * 2);     // 8B aligned (k%4==0)
        v2f b;
        b.x = Wt[(size_t)(k + half * 2 + 0) * D + n0 + l15];
        b.y = Wt[(size_t)(k + half * 2 + 1) * D + n0 + l15];
        c = __builtin_amdgcn_wmma_f32_16x16x4_f32(
                /*neg_a=*/false, a, /*neg_b=*/false, b,
                /*c_mod=*/(short)0, c, /*reuse_a=*/false, /*reuse_b=*/false);
    }

    // Store: VGPR r -> row m0 + r + half*8, col n0 + l15 (16 contiguous floats
    // per half-wave per row).
    float* orow = out + (size_t)(m0 + half * 8) * D + n0 + l15;
#pragma unroll
    for (int r = 0; r < 8; ++r) orow[(size_t)r * D] = c[r];
}

// Scalar tail for n_nodes % 16 != 0 (not hit for N_NODES=50000, kept for safety)
__global__ __launch_bounds__(128) void gemm_tail(
    const float* __restrict__ agg, const float* __restrict__ Wt,
    const float* __restrict__ bias, float* __restrict__ out,
    int row_start)
{
    int r = row_start + blockIdx.x;
    int n = threadIdx.x;
    float acc = bias[n];
    const float* a = agg + (size_t)r * D;
    for (int k = 0; k < D; ++k) acc = fmaf(a[k], Wt[(size_t)k * D + n], acc);
    out[(size_t)r * D + n] = acc;
}

// ---------------------------------------------------------------------------
extern "C" void kernel_launch(void* const* d_in, const int* in_sizes, int n_in,
                              void* d_out, int out_size, void* d_ws, size_t ws_size,
                              hipStream_t stream) {
    const float* x      = (const float*)d_in[0];
    const float* weight = (const float*)d_in[1];   // [D_IN, D_OUT] row-major
    const float* bias   = (const float*)d_in[2];
    const int*   erow   = (const int*)d_in[3];
    const int*   ecol   = (const int*)d_in[4];

    const int n_nodes = in_sizes[0] / D;           // 50000
    const int n_edges = in_sizes[3];               // 1,600,000

    float* agg = (float*)d_ws;                                   // [n_nodes, D]
    float* deg = agg + (size_t)n_nodes * D;                      // [n_nodes]

    const int zn = n_nodes * (D + 1);
    zero_ws<<<(zn + 255) / 256, 256, 0, stream>>>((float*)d_ws, zn);

    degree_count<<<(n_edges + 255) / 256, 256, 0, stream>>>(erow, deg, n_edges);
    deg_rsqrt<<<(n_nodes + 255) / 256, 256, 0, stream>>>(deg, n_nodes);

    edge_aggregate<<<(n_edges + 7) / 8, 256, 0, stream>>>(x, erow, ecol, deg,
                                                          agg, n_edges);

    const int mtiles = n_nodes / 16;               // 3125, exact for 50000
    if (mtiles > 0)
        gemm_bias_wmma<<<mtiles, 256, 0, stream>>>(agg, weight, bias,
                                                   (float*)d_out);
    const int tail = n_nodes - mtiles * 16;
    if (tail > 0)
        gemm_tail<<<tail, D, 0, stream>>>(agg, weight, bias, (float*)d_out,
                                          mtiles * 16);
}